// Applied_Hamiltonian_31215822307972
// MI455X (gfx1250) — compile-verified
//
#include <hip/hip_runtime.h>

#define DIM 2048
#define NQ  11
#define BM  128
#define BN  128
#define BK  32
#define LDA 36     // padded LDS stride, A tile: conflict-free half-wave frag reads
#define LDBP 288   // B tile k-pair-interleaved row stride: 288 % 64 == 32 bank shift
#define NSTAGE (DIM / BK)

typedef __attribute__((ext_vector_type(2))) float v2f;
typedef __attribute__((ext_vector_type(8))) float v8f;

// ---------------------------------------------------------------------------
// Kernel 1: collapse Gaussian envelope + A into per-site 2x2 complex drives.
// site[0..43]  = site_re[i][a][b]   site[44..87] = site_im[i][a][b]
// ---------------------------------------------------------------------------
__global__ __launch_bounds__(128) void site_ops_kernel(
    const float* __restrict__ A, const float* __restrict__ g_re,
    const float* __restrict__ g_im, const int* __restrict__ t_ptr,
    float* __restrict__ site) {
  __shared__ float coef[33];
  const int tid = threadIdx.x;
  const float t = (float)(*t_ptr);
  const float dt = 10.0f / 128.0f;          // T_TOTAL / M
  const float inva = 1.0f / (0.5f * dt);    // 1/a
  if (tid < 33) {
    float s = 0.0f;
    for (int m = 0; m < 128; ++m) {
      float d = (t - (float)m * dt) * inva;
      s += __expf(-d * d) * A[m * 33 + tid];
    }
    coef[tid] = s;
  }
  __syncthreads();
  if (tid < 88) {
    const int isre = (tid < 44) ? 1 : 0;
    const int j = isre ? tid : tid - 44;
    const int i = j >> 2, e = j & 3;
    const float* G = isre ? g_re : g_im;
    float s = 0.0f;
    for (int g = 0; g < 3; ++g) s += coef[g * NQ + i] * G[g * 4 + e];
    site[tid] = s;
  }
}

// ---------------------------------------------------------------------------
// Kernel 2: dense WMMA f32 GEMM, double-buffered LDS pipeline.
//   z==0: out[0] = +H0 @ Ui      z==1: out[1] = -H0 @ Ur
// 256 threads = 8 waves; block tile 128x128; wave tile 32x64 (2x4 WMMA accs).
// B tile stored k-pair interleaved so each B fragment is one ds_load_b64.
// ---------------------------------------------------------------------------
__global__ __launch_bounds__(256) void gemm_h0_kernel(
    const float* __restrict__ H0, const float* __restrict__ U,
    float* __restrict__ out) {
  __shared__ float As[2 * BM * LDA];          // 36 KB
  __shared__ float Bs[2 * (BK / 2) * LDBP];   // 36 KB

  const int tid = threadIdx.x;
  const int bx = blockIdx.x;            // N tile (16)
  const int by = blockIdx.y;            // M tile (16)
  const int z  = blockIdx.z;            // 0 -> Ui (+), 1 -> Ur (-)
  const float* Bg = U + (z == 0 ? (size_t)DIM * DIM : (size_t)0);
  float* C = out + (size_t)z * DIM * DIM;
  const float sign = (z == 0) ? 1.0f : -1.0f;

  const int lane = tid & 31;
  const int wv   = tid >> 5;
  const int wm   = (wv & 3) * 32;       // wave M offset (4 waves in M)
  const int wn   = (wv >> 2) * 64;      // wave N offset (2 waves in N)
  const int hh   = lane >> 4;           // half-wave (K split of fragments)
  const int l16  = lane & 15;

  v8f acc[2][4];
#pragma unroll
  for (int mi = 0; mi < 2; ++mi)
#pragma unroll
    for (int nj = 0; nj < 4; ++nj) acc[mi][nj] = (v8f){};

  // A staging: 128x32, 4 passes of one float4/thread
  const int arow = tid >> 3;            // 0..31
  const int acol = (tid & 7) * 4;       // 0..28
  // B staging: 16 k-pairs x 32 col-chunks, 2 passes of two float4/thread
  const int bq   = tid >> 5;            // 0..7  (row-pair base)
  const int bcc  = (tid & 31) * 4;      // 0..124 (column chunk)

  float4 ra[4], rb0[2], rb1[2];

  // prologue: fetch stage 0
#pragma unroll
  for (int p = 0; p < 4; ++p)
    ra[p] = *(const float4*)(H0 + (size_t)(by * BM + arow + 32 * p) * DIM + acol);
#pragma unroll
  for (int p = 0; p < 2; ++p) {
    const int pr = bq + 8 * p;
    rb0[p] = *(const float4*)(Bg + (size_t)(2 * pr) * DIM + bx * BN + bcc);
    rb1[p] = *(const float4*)(Bg + (size_t)(2 * pr + 1) * DIM + bx * BN + bcc);
  }
#pragma unroll
  for (int p = 0; p < 4; ++p) {
    float* a = &As[(arow + 32 * p) * LDA + acol];
    a[0] = ra[p].x; a[1] = ra[p].y; a[2] = ra[p].z; a[3] = ra[p].w;
  }
#pragma unroll
  for (int p = 0; p < 2; ++p) {
    const int pr = bq + 8 * p;
    float* b = &Bs[pr * LDBP + bcc * 2];
    b[0] = rb0[p].x; b[1] = rb1[p].x; b[2] = rb0[p].y; b[3] = rb1[p].y;
    b[4] = rb0[p].z; b[5] = rb1[p].z; b[6] = rb0[p].w; b[7] = rb1[p].w;
  }

  for (int s = 0; s < NSTAGE; ++s) {
    __syncthreads();
    const bool more = (s + 1) < NSTAGE;
    const int k1 = (s + 1) * BK;
    if (more) {   // issue next stage's global loads; overlap with WMMA below
#pragma unroll
      for (int p = 0; p < 4; ++p)
        ra[p] = *(const float4*)(H0 + (size_t)(by * BM + arow + 32 * p) * DIM + k1 + acol);
#pragma unroll
      for (int p = 0; p < 2; ++p) {
        const int pr = bq + 8 * p;
        rb0[p] = *(const float4*)(Bg + (size_t)(k1 + 2 * pr) * DIM + bx * BN + bcc);
        rb1[p] = *(const float4*)(Bg + (size_t)(k1 + 2 * pr + 1) * DIM + bx * BN + bcc);
      }
    }

    const float* Ab = &As[(s & 1) * BM * LDA];
    const float* Bb = &Bs[(s & 1) * (BK / 2) * LDBP];
#pragma unroll
    for (int kk = 0; kk < BK; kk += 4) {
      const int ka = kk + hh * 2;       // half-wave K split per ISA frag layout
      const int kp = (kk >> 1) + hh;    // interleaved k-pair row
      v2f a[2], b[4];
#pragma unroll
      for (int mi = 0; mi < 2; ++mi) {
        a[mi].x = Ab[(wm + mi * 16 + l16) * LDA + ka];
        a[mi].y = Ab[(wm + mi * 16 + l16) * LDA + ka + 1];
      }
#pragma unroll
      for (int nj = 0; nj < 4; ++nj) {
        const float* bp = &Bb[kp * LDBP + (wn + nj * 16 + l16) * 2];
        b[nj].x = bp[0];
        b[nj].y = bp[1];
      }
#pragma unroll
      for (int mi = 0; mi < 2; ++mi)
#pragma unroll
        for (int nj = 0; nj < 4; ++nj)
          acc[mi][nj] = __builtin_amdgcn_wmma_f32_16x16x4_f32(
              false, a[mi], false, b[nj], (short)0, acc[mi][nj], false, false);
    }

    if (more) {   // stage into the other LDS buffer while this one is read
      float* An = &As[((s + 1) & 1) * BM * LDA];
      float* Bn = &Bs[((s + 1) & 1) * (BK / 2) * LDBP];
#pragma unroll
      for (int p = 0; p < 4; ++p) {
        float* a = &An[(arow + 32 * p) * LDA + acol];
        a[0] = ra[p].x; a[1] = ra[p].y; a[2] = ra[p].z; a[3] = ra[p].w;
      }
#pragma unroll
      for (int p = 0; p < 2; ++p) {
        const int pr = bq + 8 * p;
        float* b = &Bn[pr * LDBP + bcc * 2];
        b[0] = rb0[p].x; b[1] = rb1[p].x; b[2] = rb0[p].y; b[3] = rb1[p].y;
        b[4] = rb0[p].z; b[5] = rb1[p].z; b[6] = rb0[p].w; b[7] = rb1[p].w;
      }
    }
  }

  // Store: C/D layout (ISA 7.12.2): VGPR v -> M = v + 8*half, N = lane&15.
  const int rbase = by * BM + wm;
  const int cbase = bx * BN + wn;
#pragma unroll
  for (int mi = 0; mi < 2; ++mi)
#pragma unroll
    for (int vv = 0; vv < 8; ++vv) {
      const int r = rbase + mi * 16 + hh * 8 + vv;
#pragma unroll
      for (int nj = 0; nj < 4; ++nj)
        C[(size_t)r * DIM + cbase + nj * 16 + l16] = sign * acc[mi][nj][vv];
    }
}

// ---------------------------------------------------------------------------
// Kernel 3: sparse H1 application. H row r has 12 nonzeros: the diagonal and
// the 11 single-bit-flip partners (site i lives at bit position 10-i).
//   out[0] += Hr@Ui + Hi@Ur        out[1] += Hi@Ui - Hr@Ur
// ---------------------------------------------------------------------------
__global__ __launch_bounds__(256) void sparse_h1_kernel(
    const float* __restrict__ U, const float* __restrict__ site,
    float* __restrict__ out) {
  __shared__ float s_re[44];
  __shared__ float s_im[44];
  const int tid = threadIdx.x;
  const int r = blockIdx.x;
  if (tid < 44) s_re[tid] = site[tid];
  else if (tid < 88) s_im[tid - 44] = site[tid];
  __syncthreads();

  const float* Ur = U;
  const float* Ui = U + (size_t)DIM * DIM;

  float hr_d = 0.0f, hi_d = 0.0f;
  float cre[NQ], cim[NQ];
  int rp[NQ];
#pragma unroll
  for (int i = 0; i < NQ; ++i) {
    const int p = 10 - i;               // bit position of site i
    const int b = (r >> p) & 1;
    hr_d += s_re[i * 4 + b * 2 + b];
    hi_d += s_im[i * 4 + b * 2 + b];
    cre[i] = s_re[i * 4 + b * 2 + (1 - b)];
    cim[i] = s_im[i * 4 + b * 2 + (1 - b)];
    rp[i] = r ^ (1 << p);
  }

  for (int c = tid; c < DIM; c += 256) {
    const float ur = Ur[(size_t)r * DIM + c];
    const float ui = Ui[(size_t)r * DIM + c];
    float accR = hr_d * ui + hi_d * ur;
    float accI = hi_d * ui - hr_d * ur;
#pragma unroll
    for (int i = 0; i < NQ; ++i) {
      const float urp = Ur[(size_t)rp[i] * DIM + c];
      const float uip = Ui[(size_t)rp[i] * DIM + c];
      accR += cre[i] * uip + cim[i] * urp;
      accI += cim[i] * uip - cre[i] * urp;
    }
    out[(size_t)r * DIM + c] += accR;
    out[(size_t)DIM * DIM + (size_t)r * DIM + c] += accI;
  }
}

// ---------------------------------------------------------------------------
extern "C" void kernel_launch(void* const* d_in, const int* in_sizes, int n_in,
                              void* d_out, int out_size, void* d_ws, size_t ws_size,
                              hipStream_t stream) {
  const float* A    = (const float*)d_in[0];   // [128, 33]
  const float* g_re = (const float*)d_in[1];   // [3, 2, 2]
  const float* g_im = (const float*)d_in[2];   // [3, 2, 2]
  const float* H0   = (const float*)d_in[3];   // [2048, 2048]
  const float* U    = (const float*)d_in[4];   // [2, 2048, 2048]
  const int*   tPtr = (const int*)d_in[5];     // scalar t
  float* out  = (float*)d_out;                 // [2, 2048, 2048]
  float* site = (float*)d_ws;                  // 88 floats of scratch

  site_ops_kernel<<<1, 128, 0, stream>>>(A, g_re, g_im, tPtr, site);

  dim3 grid(DIM / BN, DIM / BM, 2);
  gemm_h0_kernel<<<grid, 256, 0, stream>>>(H0, U, out);

  sparse_h1_kernel<<<DIM, 256, 0, stream>>>(U, site, out);
}